// GraphLearner_54142357733624
// MI455X (gfx1250) — compile-verified
//
#include <hip/hip_runtime.h>
#include <cstdint>
#include <cstddef>

#define IN_DIM 256
#define HID 128
#define NEG_SLOPE 0.2f
#define MASK_VAL -9000000000000000.0f

typedef __attribute__((ext_vector_type(2))) float v2f;
typedef __attribute__((ext_vector_type(8))) float v8f;

// ---------------------------------------------------------------------------
// Kernel A: fold the attention vectors through W once.
//   w1[d] = sum_h W[d,h] * a[h]        (d in [0,256))
//   w2[d] = sum_h W[d,h] * a[HID+h]
// This exploits (X@W)@a == X@(W@a): collapses the 4.3 GFLOP GEMM to 67 MFLOP.
// ---------------------------------------------------------------------------
__global__ __launch_bounds__(256) void fold_weights(const float* __restrict__ W,
                                                    const float* __restrict__ a,
                                                    float* __restrict__ w12) {
  const int d = threadIdx.x;                       // 0..255
  const float* __restrict__ row = W + d * HID;     // W is (IN_DIM, HID) row-major
  float acc1 = 0.f, acc2 = 0.f;
#pragma unroll 8
  for (int h = 0; h < HID; ++h) {
    const float w = row[h];
    acc1 = fmaf(w, a[h], acc1);
    acc2 = fmaf(w, a[HID + h], acc2);
  }
  w12[d]          = acc1;                          // w1
  w12[IN_DIM + d] = acc2;                          // w2
}

// ---------------------------------------------------------------------------
// Kernel B: S = X @ [w1 w2 0...0]  via V_WMMA_F32_16X16X4_F32.
// One wave32 per 16-row tile of X (X = context flattened to 65536 x 256).
// 64 accumulating WMMAs over K=256.
//
// A-matrix (16x4 f32) per-lane layout (ISA 7.12.2):
//   lanes 0-15 : M=lane,    VGPR0=K0, VGPR1=K1
//   lanes 16-31: M=lane-16, VGPR0=K2, VGPR1=K3
// B-matrix (4x16 f32), rows striped across lanes within a VGPR (mirrors C):
//   VGPR0: lanes 0-15 -> K=0 row, lanes 16-31 -> K=2 row
//   VGPR1: lanes 0-15 -> K=1 row, lanes 16-31 -> K=3 row
// C/D (16x16 f32): VGPR r: lanes 0-15 -> M=r (N=lane), lanes 16-31 -> M=r+8.
// Column 0 of D = s1 for the 16 rows, column 1 = s2.
//
// B construction is branchless: per-lane 0/1 selector floats multiplied into
// unconditionally-loaded w1/w2 pairs (avoids compiler sinking the selects
// into predicated loads with EXEC save/restore inside the WMMA loop).
// ---------------------------------------------------------------------------
__global__ __launch_bounds__(256) void row_scores(const float* __restrict__ ctx,
                                                  const float* __restrict__ w12,
                                                  float* __restrict__ s1,
                                                  float* __restrict__ s2) {
  const int wave = (int)((blockIdx.x * blockDim.x + threadIdx.x) >> 5); // tile id
  const int lane = (int)(threadIdx.x & 31);
  const int m    = lane & 15;          // A row within tile
  const int half = lane >> 4;          // 0: K-pair {0,1}, 1: K-pair {2,3}
  const int kk   = half * 2;
  const int n    = lane & 15;          // B/D column

  // Loop-invariant per-lane column selectors (0.0 or 1.0).
  const float sel1 = (n == 0) ? 1.0f : 0.0f;       // column 0 <- w1
  const float sel2 = (n == 1) ? 1.0f : 0.0f;       // column 1 <- w2

  const float* __restrict__ base = ctx + ((size_t)wave * 16 + m) * IN_DIM;
  const float* __restrict__ w1v  = w12;
  const float* __restrict__ w2v  = w12 + IN_DIM;

  v8f c = {};
#pragma unroll 4
  for (int k = 0; k < IN_DIM / 4; ++k) {
    const int K0 = 4 * k + kk;                     // even -> 8B aligned
    const v2f av = *(const v2f*)(base + K0);       // A: X[row][K0], X[row][K0+1]
    const v2f w1p = *(const v2f*)(w1v + K0);       // w1[K0], w1[K0+1]
    const v2f w2p = *(const v2f*)(w2v + K0);       // w2[K0], w2[K0+1]

    v2f bv;
    bv.x = fmaf(sel1, w1p.x, sel2 * w2p.x);        // B[K0  ][n]
    bv.y = fmaf(sel1, w1p.y, sel2 * w2p.y);        // B[K0+1][n]

    c = __builtin_amdgcn_wmma_f32_16x16x4_f32(
        /*neg_a=*/false, av, /*neg_b=*/false, bv,
        /*c_mod=*/(short)0, c, /*reuse_a=*/false, /*reuse_b=*/false);
  }

  if (n < 2) {                                     // cols 0 (s1) and 1 (s2)
    float* __restrict__ dst = (n == 0) ? s1 : s2;
    const size_t rbase = (size_t)wave * 16 + (size_t)half * 8;
#pragma unroll
    for (int r = 0; r < 8; ++r) dst[rbase + r] = c[r];
  }
}

// ---------------------------------------------------------------------------
// Kernel C: per-row masked softmax + softplus. One block per (b,i) row.
// 256 threads x 4 j's each, adj row read once (non-temporal), output written
// once (non-temporal). s1/s2 stay in L2 (512 KB vs 192 MB L2).
// This is the memory-bound part: ~512 MB traffic -> ~22 us floor @ 23.3 TB/s.
// ---------------------------------------------------------------------------
__global__ __launch_bounds__(256) void attention_rows(const int* __restrict__ adj,
                                                      const float* __restrict__ s1,
                                                      const float* __restrict__ s2,
                                                      float* __restrict__ out) {
  const int row = (int)blockIdx.x;                  // 0..65535 = b*1024 + i
  const int b   = row >> 10;
  const int t   = (int)threadIdx.x;
  const size_t rowOff = (size_t)row << 10;
  const float s1i = s1[row];
  const float* __restrict__ s2b = s2 + ((size_t)b << 10);

  float vals[4];
  float lmax = MASK_VAL;
#pragma unroll
  for (int p = 0; p < 4; ++p) {
    const int j = t + (p << 8);
    const float x = s1i + s2b[j];
    const float e = (x >= 0.f) ? x : NEG_SLOPE * x;            // leaky_relu
    const int aij = __builtin_nontemporal_load(adj + rowOff + j);
    const float v = (aij > 0) ? e : MASK_VAL;
    vals[p] = v;
    lmax = fmaxf(lmax, v);
  }

  __shared__ float red[256];
  red[t] = lmax;
  __syncthreads();
#pragma unroll
  for (int s = 128; s > 0; s >>= 1) {
    if (t < s) red[t] = fmaxf(red[t], red[t + s]);
    __syncthreads();
  }
  const float rowmax = red[0];
  __syncthreads();

  float lsum = 0.f;
#pragma unroll
  for (int p = 0; p < 4; ++p) {
    vals[p] = __expf(vals[p] - rowmax);   // masked entries -> exp(-huge) = 0
    lsum += vals[p];
  }
  red[t] = lsum;
  __syncthreads();
#pragma unroll
  for (int s = 128; s > 0; s >>= 1) {
    if (t < s) red[t] += red[t + s];
    __syncthreads();
  }
  const float inv = 1.0f / red[0];

#pragma unroll
  for (int p = 0; p < 4; ++p) {
    const int j = t + (p << 8);
    const float att = vals[p] * inv;
    const float sp = log1pf(__expf(att));            // softplus
    __builtin_nontemporal_store(sp, out + rowOff + j);
  }
}

// ---------------------------------------------------------------------------
// Launch: fold -> scores (WMMA) -> masked softmax/softplus.
// Workspace layout (floats): [0,512) w1|w2, [512, 512+65536) s1,
// [512+65536, 512+131072) s2.  ~516 KB total.
// ---------------------------------------------------------------------------
extern "C" void kernel_launch(void* const* d_in, const int* in_sizes, int n_in,
                              void* d_out, int out_size, void* d_ws, size_t ws_size,
                              hipStream_t stream) {
  const float* ctx = (const float*)d_in[0];   // (64,1024,256) f32
  const float* W   = (const float*)d_in[1];   // (256,128) f32
  const float* a   = (const float*)d_in[2];   // (256,1) f32
  const int*   adj = (const int*)d_in[3];     // (64,1024,1024) i32

  float* ws  = (float*)d_ws;
  float* w12 = ws;                  // 512 floats
  float* s1  = ws + 512;            // 65536 floats
  float* s2  = ws + 512 + 65536;    // 65536 floats
  float* out = (float*)d_out;       // (64,1024,1024) f32

  fold_weights<<<1, 256, 0, stream>>>(W, a, w12);
  // 65536 rows / 16 rows-per-wave = 4096 waves = 512 blocks x 8 waves (exact)
  row_scores<<<512, 256, 0, stream>>>(ctx, w12, s1, s2);
  attention_rows<<<64 * 1024, 256, 0, stream>>>(adj, s1, s2, out);
}